// FluxGNN1DLatent_41807211659832
// MI455X (gfx1250) — compile-verified
//
#include <hip/hip_runtime.h>

#define LATENT 32
#define HIDDEN 64
#define NCELL 8192
#define NSTEPS 64
#define NBATCH 64
#define NIF (NCELL + 1)            // 8193 interfaces per row
#define ROWSTRIDE (NSTEPS * NCELL) // batch stride in output
#define GROUPS_PER_ROW 513         // ceil(8193/16)
#define WAVES_PER_BLOCK 8

typedef __attribute__((ext_vector_type(2))) float v2f;
typedef __attribute__((ext_vector_type(8))) float v8f;

// ---- workspace float layout ----
enum {
  WS_W1S  = 0,                    // 64
  WS_W1D  = 64,                   // 64
  WS_B1   = 128,                  // 64
  WS_B2   = 192,                  // 64
  WS_B3   = 256,                  // 32
  WS_WOUT = 288,                  // 32
  WS_PW2  = 320,                  // [16][4][32][2] = 4096, WMMA B-layout
  WS_PW3  = 320 + 4096,           // [16][2][32][2] = 2048, WMMA B-layout
  WS_SCAL = 320 + 4096 + 2048,    // [0]=S, [1]=r
  WS_WTOT = WS_SCAL,              // 6464 floats preloaded to LDS
  WS_F    = 6496                  // flux scratch: NBATCH*NIF floats
};

__device__ __forceinline__ float gelu_exact(float x) {
  return 0.5f * x * (1.0f + erff(x * 0.70710678118654752440f));
}

// ---- one-time weight folding / packing ----
__global__ __launch_bounds__(256) void prep_kernel(
    const float* __restrict__ enc, const float* __restrict__ W1,
    const float* __restrict__ b1, const float* __restrict__ W2,
    const float* __restrict__ b2, const float* __restrict__ W3,
    const float* __restrict__ b3, const float* __restrict__ dt,
    const float* __restrict__ dx, float* __restrict__ ws) {
  int tid = threadIdx.x;
  if (tid < HIDDEN) {
    float s1 = 0.f, s2 = 0.f;
    for (int k = 0; k < LATENT; ++k) {
      s1 += enc[k] * W1[k * HIDDEN + tid];
      s2 += fabsf(enc[k]) * W1[(LATENT + k) * HIDDEN + tid];
    }
    ws[WS_W1S + tid] = s1;
    ws[WS_W1D + tid] = s2;
    ws[WS_B1 + tid] = b1[tid];
    ws[WS_B2 + tid] = b2[tid];
  }
  if (tid < LATENT) {
    float se = 0.f;
    for (int k = 0; k < LATENT; ++k) se += enc[k] * enc[k];
    float inv = 1.0f / (se + 1e-12f);
    ws[WS_B3 + tid] = b3[tid];
    ws[WS_WOUT + tid] = 0.25f * enc[tid] * inv;  // LATENT_FLUX_SCALE * dec
  }
  if (tid == 0) {
    float se = 0.f;
    for (int k = 0; k < LATENT; ++k) se += enc[k] * enc[k];
    ws[WS_SCAL + 0] = se / (se + 1e-12f);  // S = sum(enc*dec)
    ws[WS_SCAL + 1] = dt[0] / dx[0];       // r
  }
  // W2 into B-layout tiles: pw2[t][n][lane][q] = W2[4t+2*hi+q][16n+m]
  for (int idx = tid; idx < 16 * 4 * 32 * 2; idx += blockDim.x) {
    int q = idx & 1, lane = (idx >> 1) & 31, n = (idx >> 6) & 3, t = idx >> 8;
    int hi = lane >> 4, m = lane & 15;
    ws[WS_PW2 + idx] = W2[(4 * t + 2 * hi + q) * HIDDEN + 16 * n + m];
  }
  // W3 into B-layout tiles: pw3[t][n][lane][q] = W3[4t+2*hi+q][16n+m]
  for (int idx = tid; idx < 16 * 2 * 32 * 2; idx += blockDim.x) {
    int q = idx & 1, lane = (idx >> 1) & 31, n = (idx >> 6) & 1, t = idx >> 7;
    int hi = lane >> 4, m = lane & 15;
    ws[WS_PW3 + idx] = W3[(4 * t + 2 * hi + q) * LATENT + 16 * n + m];
  }
}

// ---- copy u0 into output slice t=0 ----
__global__ __launch_bounds__(256) void init_kernel(const float* __restrict__ u0,
                                                   float* __restrict__ out) {
  int idx = blockIdx.x * 256 + threadIdx.x;
  int b = idx >> 13;
  int j = idx & (NCELL - 1);
  out[(size_t)b * ROWSTRIDE + j] = u0[idx];
}

// ---- flux kernel: one wave computes F for 16 interfaces via f32 WMMA ----
__global__ __launch_bounds__(256) void flux_kernel(const float* __restrict__ u,
                                                   const float* __restrict__ ws,
                                                   float* __restrict__ F) {
  __shared__ float sw[WS_WTOT];
  __shared__ float sx[WAVES_PER_BLOCK][16 * 68];  // per-wave 16x64 tile, padded
  for (int i = threadIdx.x; i < WS_WTOT; i += 256) sw[i] = ws[i];
  __syncthreads();

  const int lane = threadIdx.x & 31;
  const int wv = threadIdx.x >> 5;
  const int wid = blockIdx.x * WAVES_PER_BLOCK + wv;
  const int b = wid / GROUPS_PER_ROW;
  const int g = wid - b * GROUPS_PER_ROW;
  const int j0 = g * 16;
  const int m = lane & 15;
  const int hi = lane >> 4;

  // per-interface scalar features (both half-waves redundantly load point m)
  const float* urow = u + (size_t)b * ROWSTRIDE;
  int j = j0 + m;
  int jc = j > NCELL ? NCELL : j;
  float uL = urow[jc > 0 ? jc - 1 : 0];
  float uR = urow[jc < NCELL ? jc : NCELL - 1];
  float s = uL + uR;
  float d = fabsf(uR - uL);

  // ---- layer 2: acc = gelu(layer1) @ W2 + b2, via 64 wmma_f32_16x16x4 ----
  v8f acc0, acc1, acc2, acc3;
  {
    float b20 = sw[WS_B2 + m], b21 = sw[WS_B2 + 16 + m];
    float b22 = sw[WS_B2 + 32 + m], b23 = sw[WS_B2 + 48 + m];
#pragma unroll
    for (int v = 0; v < 8; ++v) {
      acc0[v] = b20; acc1[v] = b21; acc2[v] = b22; acc3[v] = b23;
    }
  }
  const float* pw2 = &sw[WS_PW2];
#pragma unroll
  for (int t = 0; t < 16; ++t) {
    int f0 = 4 * t + 2 * hi;  // A layout: lane(m,hi) holds K = 2*hi+q of K-block t
    v2f A;
    A[0] = gelu_exact(s * sw[WS_W1S + f0] + d * sw[WS_W1D + f0] + sw[WS_B1 + f0]);
    A[1] = gelu_exact(s * sw[WS_W1S + f0 + 1] + d * sw[WS_W1D + f0 + 1] + sw[WS_B1 + f0 + 1]);
    v2f B0 = *(const v2f*)&pw2[((t * 4 + 0) * 32 + lane) * 2];
    v2f B1 = *(const v2f*)&pw2[((t * 4 + 1) * 32 + lane) * 2];
    v2f B2 = *(const v2f*)&pw2[((t * 4 + 2) * 32 + lane) * 2];
    v2f B3 = *(const v2f*)&pw2[((t * 4 + 3) * 32 + lane) * 2];
    acc0 = __builtin_amdgcn_wmma_f32_16x16x4_f32(false, A, false, B0, (short)0, acc0, false, false);
    acc1 = __builtin_amdgcn_wmma_f32_16x16x4_f32(false, A, false, B1, (short)0, acc1, false, false);
    acc2 = __builtin_amdgcn_wmma_f32_16x16x4_f32(false, A, false, B2, (short)0, acc2, false, false);
    acc3 = __builtin_amdgcn_wmma_f32_16x16x4_f32(false, A, false, B3, (short)0, acc3, false, false);
  }

  // gelu + C/D-layout -> A-layout transpose through per-wave LDS tile
  float* xb = sx[wv];
#pragma unroll
  for (int v = 0; v < 8; ++v) {
    int row = (v + 8 * hi) * 68;  // D: lane(m,hi), vgpr v => element [v+8*hi][16n+m]
    xb[row + m]      = gelu_exact(acc0[v]);
    xb[row + 16 + m] = gelu_exact(acc1[v]);
    xb[row + 32 + m] = gelu_exact(acc2[v]);
    xb[row + 48 + m] = gelu_exact(acc3[v]);
  }
  // same-wave DS ops are in-order; no barrier needed (private per-wave tile)

  // ---- layer 3: y = x2 @ W3 + b3, via 32 wmma_f32_16x16x4 ----
  v8f y0, y1;
  {
    float b30 = sw[WS_B3 + m], b31 = sw[WS_B3 + 16 + m];
#pragma unroll
    for (int v = 0; v < 8; ++v) { y0[v] = b30; y1[v] = b31; }
  }
  const float* pw3 = &sw[WS_PW3];
#pragma unroll
  for (int t = 0; t < 16; ++t) {
    v2f A = *(const v2f*)&xb[m * 68 + 4 * t + 2 * hi];
    v2f B0 = *(const v2f*)&pw3[((t * 2 + 0) * 32 + lane) * 2];
    v2f B1 = *(const v2f*)&pw3[((t * 2 + 1) * 32 + lane) * 2];
    y0 = __builtin_amdgcn_wmma_f32_16x16x4_f32(false, A, false, B0, (short)0, y0, false, false);
    y1 = __builtin_amdgcn_wmma_f32_16x16x4_f32(false, A, false, B1, (short)0, y1, false, false);
  }

  // ---- F_p = sum_k wout[k] * tanh(y[p][k]) ----
  float kw0 = sw[WS_WOUT + m];
  float kw1 = sw[WS_WOUT + 16 + m];
  float part[8];
#pragma unroll
  for (int v = 0; v < 8; ++v)
    part[v] = kw0 * tanhf(y0[v]) + kw1 * tanhf(y1[v]);

  // butterfly-reduce across the 16 lanes of each half-wave
#pragma unroll
  for (int off = 1; off < 16; off <<= 1) {
#pragma unroll
    for (int v = 0; v < 8; ++v) part[v] += __shfl_xor(part[v], off, 32);
  }

  // lane (m<8, hi) writes interface j0 + 8*hi + m
  float val = part[0];
#pragma unroll
  for (int v = 1; v < 8; ++v) val = (m == v) ? part[v] : val;
  int jj = j0 + 8 * hi + m;
  if (m < 8 && jj < NIF) F[(size_t)b * NIF + jj] = val;
}

// ---- conservative update: u_new = S*u - r*(F[j+1]-F[j]) ----
__global__ __launch_bounds__(256) void update_kernel(
    const float* __restrict__ uprev, const float* __restrict__ F,
    const float* __restrict__ ws, float* __restrict__ unew) {
  int idx = blockIdx.x * 256 + threadIdx.x;
  int b = idx >> 13;
  int j = idx & (NCELL - 1);
  float S = ws[WS_SCAL];
  float r = ws[WS_SCAL + 1];
  size_t rb = (size_t)b * ROWSTRIDE;
  const float* Fb = F + (size_t)b * NIF;
  unew[rb + j] = uprev[rb + j] * S - r * (Fb[j + 1] - Fb[j]);
}

extern "C" void kernel_launch(void* const* d_in, const int* in_sizes, int n_in,
                              void* d_out, int out_size, void* d_ws, size_t ws_size,
                              hipStream_t stream) {
  (void)in_sizes; (void)n_in; (void)out_size; (void)ws_size;
  const float* u0  = (const float*)d_in[0];
  const float* enc = (const float*)d_in[1];
  const float* W1  = (const float*)d_in[2];
  const float* b1  = (const float*)d_in[3];
  const float* W2  = (const float*)d_in[4];
  const float* b2  = (const float*)d_in[5];
  const float* W3  = (const float*)d_in[6];
  const float* b3  = (const float*)d_in[7];
  const float* dt  = (const float*)d_in[8];
  const float* dx  = (const float*)d_in[9];
  float* out = (float*)d_out;
  float* ws  = (float*)d_ws;
  float* F   = ws + WS_F;

  prep_kernel<<<1, 256, 0, stream>>>(enc, W1, b1, W2, b2, W3, b3, dt, dx, ws);
  init_kernel<<<(NBATCH * NCELL) / 256, 256, 0, stream>>>(u0, out);

  const int flux_blocks = (NBATCH * GROUPS_PER_ROW) / WAVES_PER_BLOCK;  // 4104
  for (int t = 1; t < NSTEPS; ++t) {
    const float* up = out + (size_t)(t - 1) * NCELL;
    float* un = out + (size_t)t * NCELL;
    flux_kernel<<<flux_blocks, 256, 0, stream>>>(up, ws, F);
    update_kernel<<<(NBATCH * NCELL) / 256, 256, 0, stream>>>(up, F, ws, un);
  }
}